// MaSIF_ligand_32607391711357
// MI455X (gfx1250) — compile-verified
//
#include <hip/hip_runtime.h>

typedef __attribute__((ext_vector_type(2))) float v2f;
typedef __attribute__((ext_vector_type(8))) float v8f;

#define TWO_PI_F 6.283185307179586f

// ---------------------------------------------------------------------------
// Kernel A: per (b,f) block. Gaussian aggregation (exp-bound) + fused
// [16,80]x[80,80] WMMA matmul + bias + relu + max over rotations.
// ---------------------------------------------------------------------------
__global__ __launch_bounds__(256) void conv_desc_kernel(
    const float* __restrict__ input_feat, const float* __restrict__ rho,
    const float* __restrict__ theta, const float* __restrict__ mask,
    const float* __restrict__ mu_rho, const float* __restrict__ mu_theta,
    const float* __restrict__ sigma_rho, const float* __restrict__ sigma_theta,
    const float* __restrict__ conv_W, const float* __restrict__ conv_b,
    float* __restrict__ desc)
{
    constexpr int V = 200, G = 80, F = 4;

    __shared__ float    s_rho[V], s_theta[V], s_y[V];
    __shared__ float    s_mr[G], s_mt[G], s_invsr2[G], s_invst2[G];
    __shared__ _Float16 s_gy[V][G];       // exp(rho-part) * mask * x  (fp16, values <= O(1))
    __shared__ float    s_agg[16][81];    // stride 81 -> conflict-free column reads

    const int tid = threadIdx.x;
    const int bf  = blockIdx.x;
    const int b   = bf >> 2;   // F == 4
    const int f   = bf & 3;

    // ---- stage per-b vectors and per-(f) gaussian params into LDS ----
    for (int i = tid; i < V; i += 256) {
        s_rho[i]   = rho[b * V + i];
        s_theta[i] = theta[b * V + i];
        s_y[i]     = input_feat[(b * V + i) * F + f] * mask[b * V + i];
    }
    for (int i = tid; i < G; i += 256) {
        s_mr[i] = mu_rho[f * G + i];
        s_mt[i] = mu_theta[f * G + i];
        float sr = sigma_rho[f * G + i];
        float st = sigma_theta[f * G + i];
        s_invsr2[i] = 1.0f / (sr * sr + 1e-5f);
        s_invst2[i] = 1.0f / (st * st + 1e-5f);
    }
    __syncthreads();

    // ---- rho-part gaussian (rotation invariant), folded with mask*x ----
    for (int idx = tid; idx < V * G; idx += 256) {
        int v = idx / G, g = idx - v * G;
        float d = s_rho[v] - s_mr[g];
        s_gy[v][g] = (_Float16)(__expf(-d * d * s_invsr2[g]) * s_y[v]);
    }
    __syncthreads();

    // ---- theta-part + contraction over v: thread = (r, g-lane), 5 g's each ----
    {
        const int r  = tid >> 4;
        const int gl = tid & 15;
        const float rot = (TWO_PI_F / 16.0f) * (float)r;
        float acc[5] = {0.f, 0.f, 0.f, 0.f, 0.f};
        float mt_[5], is_[5];
        #pragma unroll
        for (int j = 0; j < 5; ++j) { mt_[j] = s_mt[gl + 16*j]; is_[j] = s_invst2[gl + 16*j]; }
        for (int v = 0; v < V; ++v) {
            float th = s_theta[v] + rot;
            th -= (th >= TWO_PI_F) ? TWO_PI_F : 0.0f;   // jnp.mod(theta+ang, 2pi)
            #pragma unroll
            for (int j = 0; j < 5; ++j) {
                float d  = th - mt_[j];
                float gt = __expf(-d * d * is_[j]);     // v_exp_f32
                acc[j] += (float)s_gy[v][gl + 16*j] * gt;
            }
        }
        #pragma unroll
        for (int j = 0; j < 5; ++j) s_agg[r][gl + 16*j] = acc[j];
    }
    __syncthreads();

    // ---- WMMA: out[r, n] = relu(agg @ W_f + b_f); desc = max over r ----
    if (tid < 160) {                       // waves 0..4, one 16-wide N tile each
        const int wave = tid >> 5;
        const int lane = tid & 31;
        const int col  = lane & 15;        // A row (=r) and B column index
        const int n    = wave * 16 + col;
        const int kh   = (lane < 16) ? 0 : 2;
        const float* Wf = conv_W + f * G * G;

        v8f c0 = {}, c1 = {};              // dual accumulators: break WMMA RAW chain
        #pragma unroll
        for (int kk = 0; kk < 20; kk += 2) {   // K = 80 = 20 * 4
            int k0 = kk * 4 + kh;
            int k1 = k0 + 4;
            v2f a0, b0, a1, b1;
            a0.x = s_agg[col][k0];     a0.y = s_agg[col][k0 + 1];
            b0.x = Wf[k0 * G + n];     b0.y = Wf[(k0 + 1) * G + n];
            a1.x = s_agg[col][k1];     a1.y = s_agg[col][k1 + 1];
            b1.x = Wf[k1 * G + n];     b1.y = Wf[(k1 + 1) * G + n];
            c0 = __builtin_amdgcn_wmma_f32_16x16x4_f32(false, a0, false, b0,
                                                       (short)0, c0, false, false);
            c1 = __builtin_amdgcn_wmma_f32_16x16x4_f32(false, a1, false, b1,
                                                       (short)0, c1, false, false);
        }
        float bias = conv_b[f * G + n];
        float m = -3.4e38f;
        #pragma unroll
        for (int j = 0; j < 8; ++j)
            m = fmaxf(m, fmaxf(c0[j] + c1[j] + bias, 0.0f));  // relu then max over r
        m = fmaxf(m, __shfl_xor(m, 16));                      // rows j vs j+8 halves
        if (lane < 16) desc[b * (F * G) + f * G + n] = m;
    }
}

// ---------------------------------------------------------------------------
// Kernel B: X = relu(desc[4096,320] @ fc1_W.T[320,80] + fc1_b). One wave/tile.
// ---------------------------------------------------------------------------
__global__ __launch_bounds__(32) void fc1_kernel(
    const float* __restrict__ desc, const float* __restrict__ fc1_W,
    const float* __restrict__ fc1_b, float* __restrict__ X)
{
    constexpr int K = 320, N = 80;
    const int m0   = blockIdx.x * 16;
    const int n0   = blockIdx.y * 16;
    const int lane = threadIdx.x;
    const int idx  = lane & 15;
    const int kh   = (lane < 16) ? 0 : 2;

    v8f c0 = {}, c1 = {};
    for (int kk = 0; kk < K / 4; kk += 2) {
        int k0 = kk * 4 + kh;
        int k1 = k0 + 4;
        v2f a0, b0, a1, b1;
        a0.x = desc[(m0 + idx) * K + k0];
        a0.y = desc[(m0 + idx) * K + k0 + 1];
        b0.x = fc1_W[(n0 + idx) * K + k0];        // B[k][n] = fc1_W[n][k]
        b0.y = fc1_W[(n0 + idx) * K + k0 + 1];
        a1.x = desc[(m0 + idx) * K + k1];
        a1.y = desc[(m0 + idx) * K + k1 + 1];
        b1.x = fc1_W[(n0 + idx) * K + k1];
        b1.y = fc1_W[(n0 + idx) * K + k1 + 1];
        c0 = __builtin_amdgcn_wmma_f32_16x16x4_f32(false, a0, false, b0,
                                                   (short)0, c0, false, false);
        c1 = __builtin_amdgcn_wmma_f32_16x16x4_f32(false, a1, false, b1,
                                                   (short)0, c1, false, false);
    }
    float bias = fc1_b[n0 + idx];
    #pragma unroll
    for (int j = 0; j < 8; ++j) {
        int m = m0 + j + ((lane < 16) ? 0 : 8);
        X[m * N + n0 + idx] = fmaxf(c0[j] + c1[j] + bias, 0.0f);
    }
}

// ---------------------------------------------------------------------------
// Kernel C: split-K partials of X.T @ X  (80x80, K=4096 split 16 ways).
// grid (25, 16), one wave per block. X stays L2-resident (1.3MB << 192MB).
// ---------------------------------------------------------------------------
__global__ __launch_bounds__(32) void gram_part_kernel(
    const float* __restrict__ X, float* __restrict__ part)
{
    constexpr int N = 80;
    const int mi   = blockIdx.x / 5;
    const int nj   = blockIdx.x % 5;
    const int s    = blockIdx.y;           // K slice: 64 k-blocks of 4
    const int lane = threadIdx.x;
    const int idx  = lane & 15;
    const int kh   = (lane < 16) ? 0 : 2;

    v8f c0 = {}, c1 = {};
    const int kkBase = s * 64;
    for (int kk = 0; kk < 64; kk += 2) {
        int k0 = (kkBase + kk) * 4 + kh;
        int k1 = k0 + 4;
        v2f a0, b0, a1, b1;
        a0.x = X[k0 * N + mi * 16 + idx];          // A[m][k] = X[k][m]
        a0.y = X[(k0 + 1) * N + mi * 16 + idx];
        b0.x = X[k0 * N + nj * 16 + idx];
        b0.y = X[(k0 + 1) * N + nj * 16 + idx];
        a1.x = X[k1 * N + mi * 16 + idx];
        a1.y = X[(k1 + 1) * N + mi * 16 + idx];
        b1.x = X[k1 * N + nj * 16 + idx];
        b1.y = X[(k1 + 1) * N + nj * 16 + idx];
        c0 = __builtin_amdgcn_wmma_f32_16x16x4_f32(false, a0, false, b0,
                                                   (short)0, c0, false, false);
        c1 = __builtin_amdgcn_wmma_f32_16x16x4_f32(false, a1, false, b1,
                                                   (short)0, c1, false, false);
    }
    #pragma unroll
    for (int j = 0; j < 8; ++j) {
        int m = mi * 16 + j + ((lane < 16) ? 0 : 8);
        part[s * (N * N) + m * N + nj * 16 + idx] = c0[j] + c1[j];
    }
}

// Reduce the 16 K-slices and apply 1/B scaling.
__global__ __launch_bounds__(256) void gram_reduce_kernel(
    const float* __restrict__ part, float* __restrict__ gram)
{
    constexpr int NN = 80 * 80, S = 16;
    int idx = blockIdx.x * 256 + threadIdx.x;
    if (idx >= NN) return;
    float sum = 0.0f;
    #pragma unroll
    for (int s = 0; s < S; ++s) sum += part[s * NN + idx];
    gram[idx] = sum * (1.0f / 4096.0f);
}

// ---------------------------------------------------------------------------
// Kernel D: h = relu(gram_flat[6400] @ fc2_W.T + b2); out = h @ fco_W.T + bo
// ---------------------------------------------------------------------------
__global__ __launch_bounds__(256) void head_kernel(
    const float* __restrict__ gram, const float* __restrict__ fc2_W,
    const float* __restrict__ fc2_b, const float* __restrict__ fco_W,
    const float* __restrict__ fco_b, float* __restrict__ out)
{
    __shared__ float s_part[256];
    __shared__ float s_h[64];
    const int tid = threadIdx.x;
    const int j = tid >> 2;    // output unit 0..63
    const int p = tid & 3;     // 4-way split of K=6400

    float sum = 0.0f;
    const float* w = fc2_W + j * 6400;
    for (int i = p * 1600; i < (p + 1) * 1600; ++i)
        sum += gram[i] * w[i];
    s_part[tid] = sum;
    __syncthreads();
    if (p == 0) {
        float t = s_part[tid] + s_part[tid + 1] + s_part[tid + 2] + s_part[tid + 3];
        s_h[j] = fmaxf(t + fc2_b[j], 0.0f);
    }
    __syncthreads();
    if (tid < 7) {
        float o = fco_b[tid];
        for (int jj = 0; jj < 64; ++jj) o += s_h[jj] * fco_W[tid * 64 + jj];
        out[tid] = o;
    }
}

// ---------------------------------------------------------------------------
extern "C" void kernel_launch(void* const* d_in, const int* in_sizes, int n_in,
                              void* d_out, int out_size, void* d_ws, size_t ws_size,
                              hipStream_t stream)
{
    (void)in_sizes; (void)n_in; (void)out_size; (void)ws_size;
    const float* input_feat  = (const float*)d_in[0];
    const float* rho         = (const float*)d_in[1];
    const float* theta       = (const float*)d_in[2];
    const float* mask        = (const float*)d_in[3];
    const float* mu_rho      = (const float*)d_in[4];
    const float* mu_theta    = (const float*)d_in[5];
    const float* sigma_rho   = (const float*)d_in[6];
    const float* sigma_theta = (const float*)d_in[7];
    const float* conv_W      = (const float*)d_in[8];
    const float* conv_b      = (const float*)d_in[9];
    const float* fc1_W       = (const float*)d_in[10];
    const float* fc1_b       = (const float*)d_in[11];
    const float* fc2_W       = (const float*)d_in[12];
    const float* fc2_b       = (const float*)d_in[13];
    const float* fco_W       = (const float*)d_in[14];
    const float* fco_b       = (const float*)d_in[15];

    float* desc = (float*)d_ws;                 // [4096, 320]
    float* X    = desc + 4096 * 320;            // [4096, 80]
    float* part = X + 4096 * 80;                // [16, 80, 80] split-K partials
    float* gram = part + 16 * 80 * 80;          // [80, 80]

    conv_desc_kernel<<<4096 * 4, 256, 0, stream>>>(
        input_feat, rho, theta, mask, mu_rho, mu_theta, sigma_rho, sigma_theta,
        conv_W, conv_b, desc);
    fc1_kernel<<<dim3(256, 5), 32, 0, stream>>>(desc, fc1_W, fc1_b, X);
    gram_part_kernel<<<dim3(25, 16), 32, 0, stream>>>(X, part);
    gram_reduce_kernel<<<25, 256, 0, stream>>>(part, gram);
    head_kernel<<<1, 256, 0, stream>>>(gram, fc2_W, fc2_b, fco_W, fco_b, (float*)d_out);
}